// GNNLayer_66142496358699
// MI455X (gfx1250) — compile-verified
//
#include <hip/hip_runtime.h>

// GAT layer (PyG GATConv, H=1) for gfx1250.
// Pipeline:
//   1) k_gemm:    h = x @ W via V_WMMA_F32_16X16X4_F32 (exact f32 matrix pipe)
//   2) k_nodeatt: ai[n] = <h[n], att[:C]>, aj[n] = <h[n], att[C:]>  (wave reduce)
//   3) k_init / k_initout: zero softmax accumulators, out = bias
//   4) k_edgemax: segment max of leaky_relu(ai[dst]+aj[src]) via uint-encoded atomicMax
//   5) k_edgeexp: ex[e] = exp(alpha - max), hw global_atomic_add_f32 on denom[dst]
//   6) k_scatter: out[dst] += (ex/denom) * h[src]   (wave per edge, float4 gather,
//                 4x hardware global_atomic_add_f32 per lane — no CAS loops)

#define N_NODES 50000
#define E_EDGES 800000
#define ET      (E_EDGES + N_NODES)   // edges + self loops
#define DIN     256
#define C_OUT   128
#define NEG_SLOPE 0.2f

typedef __attribute__((ext_vector_type(2))) float v2f;
typedef __attribute__((ext_vector_type(8))) float v8f;

// Guaranteed-hardware f32 atomic add (no return): GLOBAL_ATOMIC_ADD_F32.
// s_endpgm does an implicit wait-idle, so fire-and-forget is safe; kernel
// boundaries provide the acquire for the next kernel's reads.
__device__ __forceinline__ void hw_atomic_add_f32(float* p, float v) {
  asm volatile("global_atomic_add_f32 %0, %1, off" : : "v"(p), "v"(v) : "memory");
}

// ---------------- 1) GEMM: h[N,128] = x[N,256] @ W[256,128] ----------------
// One wave per 16x16 tile. 3125 M-tiles x 8 N-tiles = 25000 tiles, exact.
__global__ __launch_bounds__(256) void k_gemm(const float* __restrict__ x,
                                              const float* __restrict__ W,
                                              float* __restrict__ h) {
  const int wave = threadIdx.x >> 5;
  const int lane = threadIdx.x & 31;
  const int tile = blockIdx.x * 8 + wave;       // [0, 25000)
  const int tM   = (tile >> 3) << 4;            // M tile base
  const int tN   = (tile & 7) << 4;             // N tile base
  const int hi   = lane >> 4;                   // half-wave selects K pair
  const int l    = lane & 15;

  const float* arow = x + (size_t)(tM + l) * DIN;
  v8f acc = (v8f)0.0f;

  for (int k = 0; k < DIN; k += 4) {
    const int ka = k + 2 * hi;                  // lanes 0-15: K=k,k+1; 16-31: K=k+2,k+3
    v2f a = *(const v2f*)(arow + ka);           // A 16x4 fragment (ISA layout)
    v2f b;
    b.x = W[(size_t)ka * C_OUT + tN + l];       // B 4x16 fragment: col = lane&15
    b.y = W[(size_t)(ka + 1) * C_OUT + tN + l];
    acc = __builtin_amdgcn_wmma_f32_16x16x4_f32(false, a, false, b,
                                                (short)0, acc, false, false);
  }
  // D layout: VGPR r -> row tM + r + 8*hi, col tN + l
  float* orow = h + (size_t)(tM + hi * 8) * C_OUT + tN + l;
#pragma unroll
  for (int r = 0; r < 8; ++r) orow[(size_t)r * C_OUT] = acc[r];
}

// ---------------- 2) per-node attention scalars ----------------
__global__ __launch_bounds__(256) void k_nodeatt(const float* __restrict__ h,
                                                 const float* __restrict__ att,
                                                 float* __restrict__ ai,
                                                 float* __restrict__ aj) {
  const int wave = threadIdx.x >> 5;
  const int lane = threadIdx.x & 31;
  const int n = blockIdx.x * 8 + wave;          // 6250 blocks * 8 = 50000 exact
  const float4 hv  = *(const float4*)(h + (size_t)n * C_OUT + lane * 4);
  const float4 av  = *(const float4*)(att + lane * 4);          // att[:C] -> h_i
  const float4 bv  = *(const float4*)(att + C_OUT + lane * 4);  // att[C:] -> h_j
  float si = hv.x*av.x + hv.y*av.y + hv.z*av.z + hv.w*av.w;
  float sj = hv.x*bv.x + hv.y*bv.y + hv.z*bv.z + hv.w*bv.w;
#pragma unroll
  for (int off = 16; off > 0; off >>= 1) {
    si += __shfl_down(si, off, 32);
    sj += __shfl_down(sj, off, 32);
  }
  if (lane == 0) { ai[n] = si; aj[n] = sj; }
}

// ---------------- 3) init ----------------
__global__ void k_init(unsigned* __restrict__ amax, float* __restrict__ denom) {
  int n = blockIdx.x * blockDim.x + threadIdx.x;
  if (n < N_NODES) { amax[n] = 0u; denom[n] = 0.0f; }
}
__global__ void k_initout(float* __restrict__ out, const float* __restrict__ bias) {
  int i = blockIdx.x * blockDim.x + threadIdx.x;
  if (i < N_NODES * C_OUT) out[i] = bias[i & (C_OUT - 1)];
}

// monotone float <-> uint encoding for atomicMax on signed floats
__device__ __forceinline__ unsigned f2key(float f) {
  unsigned b = __float_as_uint(f);
  return (b & 0x80000000u) ? ~b : (b | 0x80000000u);
}
__device__ __forceinline__ float key2f(unsigned k) {
  unsigned b = (k & 0x80000000u) ? (k & 0x7FFFFFFFu) : ~k;
  return __uint_as_float(b);
}
__device__ __forceinline__ float leaky(float a) {
  return a > 0.0f ? a : NEG_SLOPE * a;
}

// ---------------- 4) segment max over dst ----------------
__global__ void k_edgemax(const int* __restrict__ esrc, const int* __restrict__ edst,
                          const float* __restrict__ ai, const float* __restrict__ aj,
                          unsigned* __restrict__ amax) {
  int e = blockIdx.x * blockDim.x + threadIdx.x;
  if (e >= ET) return;
  int s, d;
  if (e < E_EDGES) { s = esrc[e]; d = edst[e]; } else { s = d = e - E_EDGES; }
  float a = leaky(ai[d] + aj[s]);
  atomicMax(&amax[d], f2key(a));                // native global_atomic_max_u32
}

// ---------------- 5) exp + segment sum ----------------
__global__ void k_edgeexp(const int* __restrict__ esrc, const int* __restrict__ edst,
                          const float* __restrict__ ai, const float* __restrict__ aj,
                          const unsigned* __restrict__ amax,
                          float* __restrict__ denom, float* __restrict__ ex) {
  int e = blockIdx.x * blockDim.x + threadIdx.x;
  if (e >= ET) return;
  int s, d;
  if (e < E_EDGES) { s = esrc[e]; d = edst[e]; } else { s = d = e - E_EDGES; }
  float a = leaky(ai[d] + aj[s]);
  float v = expf(a - key2f(amax[d]));
  ex[e] = v;
  hw_atomic_add_f32(&denom[d], v);
}

// ---------------- 6) weighted scatter-aggregate ----------------
// One wave per edge; lane covers 4 columns (b128 gather, 4 hw f32 atomics).
__global__ __launch_bounds__(256) void k_scatter(const int* __restrict__ esrc,
                                                 const int* __restrict__ edst,
                                                 const float* __restrict__ h,
                                                 const float* __restrict__ ex,
                                                 const float* __restrict__ denom,
                                                 float* __restrict__ out) {
  const int wave = threadIdx.x >> 5;
  const int lane = threadIdx.x & 31;
  const int e = blockIdx.x * 8 + wave;          // 106250 blocks * 8 = 850000 exact
  int s, d;
  if (e < E_EDGES) { s = esrc[e]; d = edst[e]; } else { s = d = e - E_EDGES; }
  const float coeff = ex[e] / (denom[d] + 1e-16f);
  const float4 hv = *(const float4*)(h + (size_t)s * C_OUT + lane * 4);
  float* o = out + (size_t)d * C_OUT + lane * 4;
  hw_atomic_add_f32(o + 0, hv.x * coeff);
  hw_atomic_add_f32(o + 1, hv.y * coeff);
  hw_atomic_add_f32(o + 2, hv.z * coeff);
  hw_atomic_add_f32(o + 3, hv.w * coeff);
}

extern "C" void kernel_launch(void* const* d_in, const int* in_sizes, int n_in,
                              void* d_out, int out_size, void* d_ws, size_t ws_size,
                              hipStream_t stream) {
  const float* x    = (const float*)d_in[0];   // [N, 256]
  const int*   ei   = (const int*)  d_in[1];   // [2, E]
  const float* W    = (const float*)d_in[2];   // [256, 128]
  const float* att  = (const float*)d_in[3];   // [256]
  const float* bias = (const float*)d_in[4];   // [128]
  float* out = (float*)d_out;                  // [N, 128]

  // workspace layout (floats)
  float*    h     = (float*)d_ws;                       // 6,400,000
  float*    ai    = h + (size_t)N_NODES * C_OUT;        // 50,000
  float*    aj    = ai + N_NODES;                       // 50,000
  unsigned* amax  = (unsigned*)(aj + N_NODES);          // 50,000
  float*    denom = (float*)(amax + N_NODES);           // 50,000
  float*    ex    = denom + N_NODES;                    // 850,000

  const int* esrc = ei;
  const int* edst = ei + E_EDGES;

  k_gemm   <<<3125, 256, 0, stream>>>(x, W, h);
  k_init   <<<(N_NODES + 255) / 256, 256, 0, stream>>>(amax, denom);
  k_initout<<<(N_NODES * C_OUT + 255) / 256, 256, 0, stream>>>(out, bias);
  k_nodeatt<<<N_NODES / 8, 256, 0, stream>>>(h, att, ai, aj);
  k_edgemax<<<(ET + 255) / 256, 256, 0, stream>>>(esrc, edst, ai, aj, amax);
  k_edgeexp<<<(ET + 255) / 256, 256, 0, stream>>>(esrc, edst, ai, aj, amax, denom, ex);
  k_scatter<<<ET / 8, 256, 0, stream>>>(esrc, edst, h, ex, denom, out);
}